// Net_90744069030458
// MI455X (gfx1250) — compile-verified
//
#include <hip/hip_runtime.h>

typedef __attribute__((ext_vector_type(2))) float v2f;
typedef __attribute__((ext_vector_type(8))) float v8f;

#define FDIM 128
#define HDIM 64
#define WS_STRIDE 72   // padded LDS stride for W1/W2 (bank-conflict free)
#define TS_STRIDE 68   // padded LDS stride for t strips (bank-conflict free, 8B aligned)

__device__ __forceinline__ v8f wmma_f32_16x16x4(v2f a, v2f b, v8f c) {
  // 8 args: (neg_a, A, neg_b, B, c_mod, C, reuse_a, reuse_b)
  return __builtin_amdgcn_wmma_f32_16x16x4_f32(false, a, false, b, (short)0, c,
                                               false, false);
}

// out[n x 64] = x[n x 128] @ W[128 x 64] + bias
__global__ __launch_bounds__(256) void gemm_lin_kernel(
    const float* __restrict__ x, const float* __restrict__ W,
    const float* __restrict__ bias, float* __restrict__ out, int n) {
  const int lane  = threadIdx.x & 31;
  const int wave  = threadIdx.x >> 5;
  const int c0    = (wave & 3) * 16;          // col tile
  const int r0    = blockIdx.x * 32 + (wave >> 2) * 16;
  const int l15   = lane & 15;
  const int khalf = (lane >> 4) << 1;         // 0 or 2

  int arow = r0 + l15;
  if (arow >= n) arow = n - 1;                // clamp: keep EXEC full for WMMA

  const float bval = bias[c0 + l15];
  v8f c;
#pragma unroll
  for (int r = 0; r < 8; ++r) c[r] = bval;

#pragma unroll
  for (int k = 0; k < FDIM; k += 4) {
    const int kb = k + khalf;
    v2f a;
    a.x = x[(size_t)arow * FDIM + kb];
    a.y = x[(size_t)arow * FDIM + kb + 1];
    v2f b;
    b.x = W[(size_t)kb * HDIM + c0 + l15];
    b.y = W[(size_t)(kb + 1) * HDIM + c0 + l15];
    c = wmma_f32_16x16x4(a, b, c);
  }

  const int rbase = r0 + ((lane >> 4) << 3);
#pragma unroll
  for (int r = 0; r < 8; ++r) {
    const int row = rbase + r;
    if (row < n) out[(size_t)row * HDIM + c0 + l15] = c[r];
  }
}

// m[dst] += h[src] * attr  (segment_sum); 16 threads per edge, float4 each
__global__ __launch_bounds__(256) void scatter_kernel(
    const float* __restrict__ h, const int* __restrict__ src,
    const int* __restrict__ dst, const float* __restrict__ attr,
    float* __restrict__ m, int E) {
  const unsigned t = blockIdx.x * 256u + threadIdx.x;
  const unsigned e = t >> 4;
  if (e >= (unsigned)E) return;
  const int cc = (t & 15u) << 2;
  const int s = src[e];
  const int d = dst[e];
  const float a = attr[e];
  const float4 hv = *(const float4*)(h + (size_t)s * HDIM + cc);
  float* mp = m + (size_t)d * HDIM + cc;
  atomicAdd(mp + 0, hv.x * a);
  atomicAdd(mp + 1, hv.y * a);
  atomicAdd(mp + 2, hv.z * a);
  atomicAdd(mp + 3, hv.w * a);
}

// out = relu(m @ W1 + b1) @ W2 + b2 (+ res);  8 waves, each owns a 16x64 strip
__global__ __launch_bounds__(256) void mlp_fused_kernel(
    const float* __restrict__ m, const float* __restrict__ W1,
    const float* __restrict__ b1, const float* __restrict__ W2,
    const float* __restrict__ b2, const float* __restrict__ res,
    float* __restrict__ out, int n) {
  __shared__ float W1s[HDIM * WS_STRIDE];
  __shared__ float W2s[HDIM * WS_STRIDE];
  __shared__ float ts[8 * 16 * TS_STRIDE];

  for (int i = threadIdx.x; i < HDIM * HDIM; i += 256) {
    const int row = i >> 6, col = i & 63;
    W1s[row * WS_STRIDE + col] = W1[i];
    W2s[row * WS_STRIDE + col] = W2[i];
  }
  __syncthreads();

  const int lane  = threadIdx.x & 31;
  const int wave  = threadIdx.x >> 5;
  const int l15   = lane & 15;
  const int khalf = (lane >> 4) << 1;
  const int r0    = (blockIdx.x * 8 + wave) * 16;
  float* tw = ts + wave * 16 * TS_STRIDE;

  int arow = r0 + l15;
  if (arow >= n) arow = n - 1;                // clamp: keep EXEC full for WMMA

  v8f acc[4];
#pragma unroll
  for (int t = 0; t < 4; ++t) {
    const float bv = b1[t * 16 + l15];
#pragma unroll
    for (int r = 0; r < 8; ++r) acc[t][r] = bv;
  }

  // GEMM1: m(16x64) @ W1
#pragma unroll
  for (int k = 0; k < HDIM; k += 4) {
    const int kb = k + khalf;
    v2f a;
    a.x = m[(size_t)arow * HDIM + kb];
    a.y = m[(size_t)arow * HDIM + kb + 1];
#pragma unroll
    for (int t = 0; t < 4; ++t) {
      v2f b;
      b.x = W1s[kb * WS_STRIDE + t * 16 + l15];
      b.y = W1s[(kb + 1) * WS_STRIDE + t * 16 + l15];
      acc[t] = wmma_f32_16x16x4(a, b, acc[t]);
    }
  }

  // relu -> t strip in LDS (private to this wave; same-wave DS ordering suffices)
  const int rbase = (lane >> 4) << 3;
#pragma unroll
  for (int t = 0; t < 4; ++t) {
#pragma unroll
    for (int r = 0; r < 8; ++r) {
      const float v = acc[t][r];
      tw[(rbase + r) * TS_STRIDE + t * 16 + l15] = v > 0.f ? v : 0.f;
    }
  }

  // GEMM2: t(16x64) @ W2
#pragma unroll
  for (int t = 0; t < 4; ++t) {
    const float bv = b2[t * 16 + l15];
#pragma unroll
    for (int r = 0; r < 8; ++r) acc[t][r] = bv;
  }
#pragma unroll
  for (int k = 0; k < HDIM; k += 4) {
    const int kb = k + khalf;
    v2f a;
    a.x = tw[l15 * TS_STRIDE + kb];
    a.y = tw[l15 * TS_STRIDE + kb + 1];
#pragma unroll
    for (int t = 0; t < 4; ++t) {
      v2f b;
      b.x = W2s[kb * WS_STRIDE + t * 16 + l15];
      b.y = W2s[(kb + 1) * WS_STRIDE + t * 16 + l15];
      acc[t] = wmma_f32_16x16x4(a, b, acc[t]);
    }
  }

  const int rowbase = r0 + rbase;
#pragma unroll
  for (int t = 0; t < 4; ++t) {
#pragma unroll
    for (int r = 0; r < 8; ++r) {
      const int row = rowbase + r;
      if (row < n) {
        float v = acc[t][r];
        if (res) v += res[(size_t)row * HDIM + t * 16 + l15];
        out[(size_t)row * HDIM + t * 16 + l15] = v;
      }
    }
  }
}

extern "C" void kernel_launch(void* const* d_in, const int* in_sizes, int n_in,
                              void* d_out, int out_size, void* d_ws,
                              size_t ws_size, hipStream_t stream) {
  const float* x     = (const float*)d_in[0];
  const int*   eidx  = (const int*)d_in[1];
  const float* attr  = (const float*)d_in[2];
  const float* lin_W = (const float*)d_in[3];
  const float* lin_b = (const float*)d_in[4];
  const float* W1    = (const float*)d_in[5];
  const float* b1    = (const float*)d_in[6];
  const float* W2    = (const float*)d_in[7];
  const float* b2    = (const float*)d_in[8];

  const int N = in_sizes[0] / FDIM;     // 50000
  const int E = in_sizes[2];            // 800000
  const int* src = eidx;
  const int* dst = eidx + E;

  float* scats = (float*)d_out;                       // (3, N, 64)
  float* outs  = scats + (size_t)3 * N * HDIM;        // (3, N, 64)

  float* h0   = (float*)d_ws;                         // N*64 floats
  float* mbuf = h0 + (size_t)N * HDIM;                // N*64 floats

  // h0 = x @ lin_W + lin_b
  gemm_lin_kernel<<<(N + 31) / 32, 256, 0, stream>>>(x, lin_W, lin_b, h0, N);

  const float* hprev = h0;
  for (int i = 0; i < 3; ++i) {
    const float* W1i = W1 + (size_t)i * HDIM * HDIM;
    const float* b1i = b1 + (size_t)i * HDIM;
    const float* W2i = W2 + (size_t)i * HDIM * HDIM;
    const float* b2i = b2 + (size_t)i * HDIM;
    float* houti = outs + (size_t)i * N * HDIM;
    float* scati = scats + (size_t)i * N * HDIM;

    // h = h + conv(h, i)
    hipMemsetAsync(mbuf, 0, (size_t)N * HDIM * sizeof(float), stream);
    scatter_kernel<<<(E * 16 + 255) / 256, 256, 0, stream>>>(hprev, src, dst,
                                                             attr, mbuf, E);
    mlp_fused_kernel<<<(N + 127) / 128, 256, 0, stream>>>(mbuf, W1i, b1i, W2i,
                                                          b2i, hprev, houti, N);

    // scats[i] = conv(h, i)
    hipMemsetAsync(mbuf, 0, (size_t)N * HDIM * sizeof(float), stream);
    scatter_kernel<<<(E * 16 + 255) / 256, 256, 0, stream>>>(houti, src, dst,
                                                             attr, mbuf, E);
    mlp_fused_kernel<<<(N + 127) / 128, 256, 0, stream>>>(
        mbuf, W1i, b1i, W2i, b2i, nullptr, scati, N);

    hprev = houti;
  }
}